// ModalityMoE_53188874994144
// MI455X (gfx1250) — compile-verified
//
#include <hip/hip_runtime.h>
#include <hip/hip_bf16.h>
#include <math.h>

// Problem constants (from reference)
#define B_  8
#define L_  384
#define D_  1024
#define E_  4
#define H_  4096
#define LHD 128   // head length
#define LWR 128   // wrist length (prop = 128)

// GEMM tiling: 128x128 macro tile, 8 waves (4M x 2N), wave tile 32x64 (2x4 WMMA)
#define BM 128
#define BN 128
#define BK 32
#define XSP 36    // padded LDS stride (floats) for A tiles, row-major [m][k]
#define BNP 136   // padded LDS stride (floats) for B tiles, row-major [k][n]
                  // 2*136 = 16 mod 64 -> lane halves hit disjoint bank windows

typedef float v2f __attribute__((ext_vector_type(2)));
typedef float v8f __attribute__((ext_vector_type(8)));

__device__ __forceinline__ float gelu_tanh(float x) {
    float x3 = x * x * x;
    return 0.5f * x * (1.0f + tanhf(0.7978845608028654f * (x + 0.044715f * x3)));
}

__device__ __forceinline__ v8f wmma4(v2f a, v2f b, v8f c) {
    // V_WMMA_F32_16X16X4_F32 : D = A(16x4) * B(4x16) + C(16x16), fp32 matrix pipe
    return __builtin_amdgcn_wmma_f32_16x16x4_f32(false, a, false, b, (short)0, c,
                                                 false, false);
}

// ---- CDNA5 async global->LDS copy (ASYNCcnt path), per-lane 16B -------------
__device__ __forceinline__ void async_b128(const float* g, const float* lds_ptr) {
    // low 32 bits of a flat shared pointer are the LDS byte address (ISA 10.2)
    unsigned int off = (unsigned int)(unsigned long long)lds_ptr;
    asm volatile("global_load_async_to_lds_b128 %0, %1, off"
                 :: "v"(off), "v"(g) : "memory");
}
__device__ __forceinline__ void wait_async_8() {
    asm volatile("s_wait_asynccnt 0x8" ::: "memory");
}
__device__ __forceinline__ void wait_async_0() {
    asm volatile("s_wait_asynccnt 0x0" ::: "memory");
}

// ---------------------------------------------------------------------------
// Gating: aggregates over L, logits, silu-mod, softmax, top-2 -> (idx, wt)
// ---------------------------------------------------------------------------
__global__ __launch_bounds__(256) void gate_kernel(
    const float* __restrict__ ctx, const float* __restrict__ tc,
    const float* __restrict__ gW, const float* __restrict__ gb,
    const float* __restrict__ tW, const float* __restrict__ tb,
    int* __restrict__ eidx, float* __restrict__ ewt)
{
    const int b   = blockIdx.x;
    const int tid = threadIdx.x;
    __shared__ float slog[E_];
    __shared__ float smod[2 * E_];
    if (tid < E_)     slog[tid] = 0.0f;
    if (tid < 2 * E_) smod[tid] = 0.0f;
    __syncthreads();

    float lac[E_]     = {0.f, 0.f, 0.f, 0.f};
    float mac[2 * E_] = {0.f, 0.f, 0.f, 0.f, 0.f, 0.f, 0.f, 0.f};
    const float* xb = ctx + (size_t)b * L_ * D_;

    for (int d = tid; d < D_; d += 256) {
        float sh = 0.f, sw = 0.f, sp = 0.f;
        for (int l = 0;           l < LHD;        ++l) sh += xb[l * D_ + d];
        for (int l = LHD;         l < LHD + LWR;  ++l) sw += xb[l * D_ + d];
        for (int l = LHD + LWR;   l < L_;         ++l) sp += xb[l * D_ + d];
        const float full = (sh + sw + sp) * (1.0f / (float)L_);
        const float hp   = (sh + sp) * (1.0f / 256.0f);
        const float wp   = (sw + sp) * (1.0f / 256.0f);
        #pragma unroll
        for (int e = 0; e < E_; ++e) {
            const float* g = gW + (size_t)e * 3 * D_;
            lac[e] += full * g[d] + hp * g[D_ + d] + wp * g[2 * D_ + d];
        }
        const float t  = tc[b * D_ + d];
        const float sl = t / (1.0f + __expf(-t));   // silu
        #pragma unroll
        for (int j = 0; j < 2 * E_; ++j) mac[j] += sl * tW[(size_t)j * D_ + d];
    }
    #pragma unroll
    for (int e = 0; e < E_; ++e)     atomicAdd(&slog[e], lac[e]);
    #pragma unroll
    for (int j = 0; j < 2 * E_; ++j) atomicAdd(&smod[j], mac[j]);
    __syncthreads();

    if (tid == 0) {
        float lg[E_], sc[E_];
        #pragma unroll
        for (int e = 0; e < E_; ++e) {
            const float scale = smod[e] + tb[e];
            const float shift = smod[E_ + e] + tb[E_ + e];
            lg[e] = (slog[e] + gb[e]) * (1.0f + scale) + shift;
        }
        float mx = lg[0];
        for (int e = 1; e < E_; ++e) mx = fmaxf(mx, lg[e]);
        float den = 0.f;
        for (int e = 0; e < E_; ++e) { sc[e] = __expf(lg[e] - mx); den += sc[e]; }
        for (int e = 0; e < E_; ++e) sc[e] /= den;
        int i0 = 0;
        for (int e = 1; e < E_; ++e) if (sc[e] > sc[i0]) i0 = e;
        int i1 = -1;
        for (int e = 0; e < E_; ++e) {
            if (e == i0) continue;
            if (i1 < 0 || sc[e] > sc[i1]) i1 = e;
        }
        const float inv = 1.0f / (sc[i0] + sc[i1] + 1e-8f);
        eidx[b * 2 + 0] = i0;           eidx[b * 2 + 1] = i1;
        ewt [b * 2 + 0] = sc[i0] * inv; ewt [b * 2 + 1] = sc[i1] * inv;
    }
}

// ---------------------------------------------------------------------------
// Double-buffered, async-staged fp32 WMMA GEMM core.
// A: (M x KDIM) row-major stride LDA; B: (KDIM x N) row-major stride LDB.
// Block tile 128x128, 8 waves 4(M)x2(N), wave tile 32x64 = 2x4 WMMA 16x16.
// ---------------------------------------------------------------------------
template<int LDA, int LDB, int KDIM>
__device__ __forceinline__ void gemm_core(const float* __restrict__ A,
                                          const float* __restrict__ Bm,
                                          float* lx, float* lw,
                                          int m0, int n0, v8f acc[2][4])
{
    const int tid  = threadIdx.x;
    const int lane = tid & 31, wid = tid >> 5;
    const int wm = wid & 3, wn = wid >> 2;       // wave grid 4(M) x 2(N)
    const int r16 = lane & 15, half = lane >> 4;

    const int xrow = tid >> 3, xc4 = (tid & 7) * 4;    // A stage: (+p*32) rows
    const int wk   = tid >> 5, wc4 = (tid & 31) * 4;   // B stage: (+p*8) k-rows

    const int NCH = KDIM / BK;

    // prefetch chunk 0 into buffer 0 (8 async b128 per thread per chunk)
    {
        #pragma unroll
        for (int p = 0; p < 4; ++p)
            async_b128(A + (size_t)(m0 + p * 32 + xrow) * LDA + xc4,
                       lx + (p * 32 + xrow) * XSP + xc4);
        #pragma unroll
        for (int p = 0; p < 4; ++p)
            async_b128(Bm + (size_t)(p * 8 + wk) * LDB + n0 + wc4,
                       lw + (p * 8 + wk) * BNP + wc4);
    }

    for (int i = 0; i < NCH; ++i) {
        float* cx = lx + (i & 1) * (BM * XSP);
        float* cw = lw + (i & 1) * (BK * BNP);
        if (i + 1 < NCH) {
            const int kc = (i + 1) * BK;
            float* bx = lx + ((i + 1) & 1) * (BM * XSP);
            float* bw = lw + ((i + 1) & 1) * (BK * BNP);
            #pragma unroll
            for (int p = 0; p < 4; ++p)
                async_b128(A + (size_t)(m0 + p * 32 + xrow) * LDA + kc + xc4,
                           bx + (p * 32 + xrow) * XSP + xc4);
            #pragma unroll
            for (int p = 0; p < 4; ++p)
                async_b128(Bm + (size_t)(kc + p * 8 + wk) * LDB + n0 + wc4,
                           bw + (p * 8 + wk) * BNP + wc4);
            wait_async_8();   // chunk i complete; chunk i+1 still in flight
        } else {
            wait_async_0();
        }
        __syncthreads();

        #pragma unroll
        for (int kk = 0; kk < BK; kk += 4) {
            const int ko = kk + 2 * half;   // lanes 0-15: K{0,1}; 16-31: K{2,3}
            v2f a0 = *(const v2f*)&cx[(wm * 32 + r16)      * XSP + ko];
            v2f a1 = *(const v2f*)&cx[(wm * 32 + 16 + r16) * XSP + ko];
            const float* bp = &cw[ko * BNP + wn * 64 + r16];
            #pragma unroll
            for (int nt = 0; nt < 4; ++nt) {
                v2f bv; bv.x = bp[nt * 16]; bv.y = bp[nt * 16 + BNP]; // ds_load_2addr_b32
                acc[0][nt] = wmma4(a0, bv, acc[0][nt]);
                acc[1][nt] = wmma4(a1, bv, acc[1][nt]);
            }
        }
        __syncthreads();
    }
}

// ---------------------------------------------------------------------------
// GEMM1: h = gelu(X(LxD) @ W1(DxH) + b1)  for slots {expert0, expert1, shared}
// ---------------------------------------------------------------------------
__global__ __launch_bounds__(256) void moe_gemm1(
    const float* __restrict__ ctx,
    const float* __restrict__ eW1, const float* __restrict__ eb1,
    const float* __restrict__ sW1, const float* __restrict__ sb1,
    const int* __restrict__ eidx,
    float* __restrict__ hbuf, int b_base)
{
    __shared__ float lx[2 * BM * XSP];
    __shared__ float lw[2 * BK * BNP];

    const int tid   = threadIdx.x;
    const int s     = blockIdx.y;           // slot: 0,1 experts; 2 shared
    const int bz    = blockIdx.z;           // local batch
    const int b     = b_base + bz;
    const int tileM = blockIdx.x % (L_ / BM);
    const int tileN = blockIdx.x / (L_ / BM);
    const int m0 = tileM * BM, n0 = tileN * BN;

    const float *W1, *bias1;
    if (s < 2) {
        const int e = eidx[b * 2 + s];
        W1 = eW1 + (size_t)e * D_ * H_;  bias1 = eb1 + (size_t)e * H_;
    } else {
        W1 = sW1;                        bias1 = sb1;
    }
    const float* X = ctx + (size_t)b * L_ * D_;

    const int lane = tid & 31, wid = tid >> 5;
    const int wm = wid & 3, wn = wid >> 2;
    const int r16 = lane & 15, half = lane >> 4;

    const v8f z = {0.f, 0.f, 0.f, 0.f, 0.f, 0.f, 0.f, 0.f};
    v8f acc[2][4] = {{z, z, z, z}, {z, z, z, z}};

    gemm_core<D_, H_, D_>(X, W1, lx, lw, m0, n0, acc);

    float* hb = hbuf + (size_t)(bz * 3 + s) * L_ * H_;
    #pragma unroll
    for (int mt = 0; mt < 2; ++mt)
        #pragma unroll
        for (int nt = 0; nt < 4; ++nt) {
            const int col = n0 + wn * 64 + nt * 16 + r16;
            const float bv = bias1[col];
            #pragma unroll
            for (int i = 0; i < 8; ++i) {
                const int row = m0 + wm * 32 + mt * 16 + half * 8 + i;
                hb[(size_t)row * H_ + col] = gelu_tanh(acc[mt][nt][i] + bv);
            }
        }
}

// ---------------------------------------------------------------------------
// GEMM2: out = sum_slots wgt*mask*( h(LxH) @ W2(HxD) + b2 )
// Slot combine is a read-modify-write epilogue on out (each element owned by
// exactly one block; slot 0 stores, slots 1-2 accumulate).
// ---------------------------------------------------------------------------
__global__ __launch_bounds__(256) void moe_gemm2(
    const float* __restrict__ hbuf,
    const float* __restrict__ eW2, const float* __restrict__ eb2,
    const float* __restrict__ sW2, const float* __restrict__ sb2,
    const int* __restrict__ eidx, const float* __restrict__ ewt,
    float* __restrict__ out, int b_base)
{
    __shared__ float lx[2 * BM * XSP];
    __shared__ float lw[2 * BK * BNP];

    const int tid   = threadIdx.x;
    const int bz    = blockIdx.y;
    const int b     = b_base + bz;
    const int tileM = blockIdx.x % (L_ / BM);
    const int tileN = blockIdx.x / (L_ / BM);
    const int m0 = tileM * BM, n0 = tileN * BN;

    const int lane = tid & 31, wid = tid >> 5;
    const int wm = wid & 3, wn = wid >> 2;
    const int r16 = lane & 15, half = lane >> 4;

    const v8f z = {0.f, 0.f, 0.f, 0.f, 0.f, 0.f, 0.f, 0.f};
    float* ob = out + (size_t)b * L_ * D_;

    for (int s = 0; s < 3; ++s) {
        const float *W2, *bias2; float wgt; int e;
        if (s < 2) {
            e = eidx[b * 2 + s];
            W2 = eW2 + (size_t)e * H_ * D_;  bias2 = eb2 + (size_t)e * D_;
            wgt = ewt[b * 2 + s];
        } else {
            e = -1;  W2 = sW2;  bias2 = sb2;  wgt = 1.0f;
        }
        const float* A = hbuf + (size_t)(bz * 3 + s) * L_ * H_;

        v8f acc[2][4] = {{z, z, z, z}, {z, z, z, z}};
        gemm_core<H_, D_, H_>(A, W2, lx, lw, m0, n0, acc);

        // weight * mask * (acc + bias) -> out (RMW across slots)
        #pragma unroll
        for (int mt = 0; mt < 2; ++mt)
            #pragma unroll
            for (int nt = 0; nt < 4; ++nt) {
                const int col = n0 + wn * 64 + nt * 16 + r16;
                const float bv = bias2[col];
                #pragma unroll
                for (int i = 0; i < 8; ++i) {
                    const int l = m0 + wm * 32 + mt * 16 + half * 8 + i;
                    float m = 1.0f;
                    if (e == 1 && l >= LHD && l < LHD + LWR) m = 0.0f;  // expert1: wrist masked
                    if (e == 2 && l < LHD)                   m = 0.0f;  // expert2: head masked
                    const float contrib = wgt * m * (acc[mt][nt][i] + bv);
                    float* p = &ob[(size_t)l * D_ + col];
                    *p = (s == 0) ? contrib : (*p + contrib);
                }
            }
    }
}

// ---------------------------------------------------------------------------
extern "C" void kernel_launch(void* const* d_in, const int* in_sizes, int n_in,
                              void* d_out, int out_size, void* d_ws, size_t ws_size,
                              hipStream_t stream)
{
    const float* ctx = (const float*)d_in[0];
    const float* tc  = (const float*)d_in[1];
    const float* gW  = (const float*)d_in[2];
    const float* gb  = (const float*)d_in[3];
    const float* tW  = (const float*)d_in[4];
    const float* tb  = (const float*)d_in[5];
    const float* eW1 = (const float*)d_in[6];
    const float* eb1 = (const float*)d_in[7];
    const float* eW2 = (const float*)d_in[8];
    const float* eb2 = (const float*)d_in[9];
    const float* sW1 = (const float*)d_in[10];
    const float* sb1 = (const float*)d_in[11];
    const float* sW2 = (const float*)d_in[12];
    const float* sb2 = (const float*)d_in[13];
    float* out = (float*)d_out;

    char*  ws   = (char*)d_ws;
    int*   eidx = (int*)ws;                    // [B][2]
    float* ewt  = (float*)(ws + 256);          // [B][2]
    float* hbuf = (float*)(ws + 512);          // [nb][3][L][H]

    const size_t per_b = (size_t)3 * L_ * H_ * sizeof(float);
    size_t cap = (ws_size > 512) ? (ws_size - 512) / per_b : 0;
    int nb_max = (int)(cap < 1 ? 1 : (cap > (size_t)B_ ? (size_t)B_ : cap));

    gate_kernel<<<B_, 256, 0, stream>>>(ctx, tc, gW, gb, tW, tb, eidx, ewt);

    for (int b0 = 0; b0 < B_; b0 += nb_max) {
        const int nb = (B_ - b0 < nb_max) ? (B_ - b0) : nb_max;
        dim3 g1((L_ / BM) * (H_ / BN), 3, nb);
        moe_gemm1<<<g1, 256, 0, stream>>>(ctx, eW1, eb1, sW1, sb1, eidx, hbuf, b0);
        dim3 g2((L_ / BM) * (D_ / BN), nb);
        moe_gemm2<<<g2, 256, 0, stream>>>(hbuf, eW2, eb2, sW2, sb2, eidx, ewt, out, b0);
    }
}